// Cell_34832184770684
// MI455X (gfx1250) — compile-verified
//
#include <hip/hip_runtime.h>
#include <math.h>

// ---------------------------------------------------------------------------
// SPD DARTS cell on MI455X (gfx1250).
// - All congruence products on the fp32 matrix pipe: V_WMMA_F32_16X16X4_F32.
// - Global->LDS staging via GLOBAL_LOAD_ASYNC_TO_LDS_B128 (ASYNCcnt), with
//   double-buffered channel loops in the dominant bimap kernel.
// - Eigendecompositions: workgroup-per-matrix parallel cyclic Jacobi in LDS.
// ---------------------------------------------------------------------------

typedef __attribute__((ext_vector_type(2))) float v2f;
typedef __attribute__((ext_vector_type(8))) float v8f;

#define DEVINL __device__ __forceinline__

// D = A(16x4) * B(4x16) + C, full fp32 WMMA
DEVINL v8f wmma_k4(v2f a, v2f b, v8f c) {
  return __builtin_amdgcn_wmma_f32_16x16x4_f32(false, a, false, b, (short)0, c,
                                               false, false);
}

// Async global->LDS copy of 16 bytes (VGPR-bypassing, tracked by ASYNCcnt).
// ISA 10.2: the LDS byte offset is taken from the low 32 bits of the address.
DEVINL void async_ld_b128(float* lds_dst, const float* gsrc) {
  unsigned lo = (unsigned)(uintptr_t)lds_dst;
  asm volatile("global_load_async_to_lds_b128 %0, %1, off"
               :: "v"(lo), "v"(gsrc) : "memory");
}
DEVINL void wait_async0() {
  asm volatile("s_wait_asynccnt 0x0" ::: "memory");
}

// A-fragment: 16x4 block at (m0,k0) of row-major LDS matrix, leading dim ld.
// ISA layout: lanes 0-15 M=lane, VGPR v holds K=v (+2 for lanes 16-31).
DEVINL v2f ldsA(const float* m, int ld, int m0, int k0, int lane) {
  int h = lane >> 4, r = lane & 15;
  v2f a;
  a.x = m[(m0 + r) * ld + k0 + 2 * h];
  a.y = m[(m0 + r) * ld + k0 + 2 * h + 1];
  return a;
}
// B-fragment: 4x16 block at (k0,n0), row-major source.
DEVINL v2f ldsB(const float* m, int ld, int k0, int n0, int lane) {
  int h = lane >> 4, r = lane & 15;
  v2f b;
  b.x = m[(k0 + 2 * h) * ld + n0 + r];
  b.y = m[(k0 + 2 * h + 1) * ld + n0 + r];
  return b;
}
// B-fragment of M^T where LDS holds M row-major: B[k][n] = M[n][k].
DEVINL v2f ldsBT(const float* m, int ld, int k0, int n0, int lane) {
  int h = lane >> 4, r = lane & 15;
  v2f b;
  b.x = m[(n0 + r) * ld + k0 + 2 * h];
  b.y = m[(n0 + r) * ld + k0 + 2 * h + 1];
  return b;
}
// C/D store: VGPR v -> row m0+v (+8 for lanes 16-31), col n0+lane%16.
DEVINL void stC(float* m, int ld, int m0, int n0, int lane, v8f c) {
  int h = lane >> 4, r = lane & 15;
#pragma unroll
  for (int v = 0; v < 8; ++v) m[(m0 + v + 8 * h) * ld + n0 + r] = c[v];
}

// ---------------------------------------------------------------------------
// Preprocess bimap: Y[b,o] = sym(sum_c W[o,c]·X[b,c]·W[o,c]^T) + 1e-3·I, n=64.
// One workgroup (8 waves) per output matrix; async LDS staging per channel.
// ---------------------------------------------------------------------------
template <int CI>
__global__ void __launch_bounds__(256)
bimap64_kernel(const float* __restrict__ X, const float* __restrict__ W,
               float* __restrict__ Y, int Co) {
  __shared__ float Wm[64 * 64];
  __shared__ float Xm[64 * 64];
  __shared__ float Tm[64 * 64];
  const int b = blockIdx.x / Co, o = blockIdx.x % Co;
  const int tid = threadIdx.x, wave = tid >> 5, lane = tid & 31;
  v8f acc0 = {}, acc1 = {};
  for (int c = 0; c < CI; ++c) {
    const float* Wp = W + (size_t)(o * CI + c) * 4096;
    const float* Xp = X + (size_t)(b * CI + c) * 4096;
    for (int i = tid; i < 1024; i += 256) {   // 1024 x b128 chunks per matrix
      async_ld_b128(&Wm[i * 4], Wp + i * 4);
      async_ld_b128(&Xm[i * 4], Xp + i * 4);
    }
    wait_async0();
    __syncthreads();
    // stage 1: T = W * X
#pragma unroll
    for (int t = 0; t < 2; ++t) {
      int tile = wave * 2 + t, m0 = (tile >> 2) * 16, n0 = (tile & 3) * 16;
      v8f cc = {};
#pragma unroll
      for (int k0 = 0; k0 < 64; k0 += 4)
        cc = wmma_k4(ldsA(Wm, 64, m0, k0, lane), ldsB(Xm, 64, k0, n0, lane), cc);
      stC(Tm, 64, m0, n0, lane, cc);
    }
    __syncthreads();
    // stage 2: acc += T * W^T
    {
      int tile = wave * 2 + 0, m0 = (tile >> 2) * 16, n0 = (tile & 3) * 16;
#pragma unroll
      for (int k0 = 0; k0 < 64; k0 += 4)
        acc0 = wmma_k4(ldsA(Tm, 64, m0, k0, lane), ldsBT(Wm, 64, k0, n0, lane), acc0);
      tile = wave * 2 + 1; m0 = (tile >> 2) * 16; n0 = (tile & 3) * 16;
#pragma unroll
      for (int k0 = 0; k0 < 64; k0 += 4)
        acc1 = wmma_k4(ldsA(Tm, 64, m0, k0, lane), ldsBT(Wm, 64, k0, n0, lane), acc1);
    }
    __syncthreads();
  }
  // symmetrize + jitter via LDS
  {
    int tile = wave * 2 + 0;
    stC(Xm, 64, (tile >> 2) * 16, (tile & 3) * 16, lane, acc0);
    tile = wave * 2 + 1;
    stC(Xm, 64, (tile >> 2) * 16, (tile & 3) * 16, lane, acc1);
  }
  __syncthreads();
  float* Yp = Y + (size_t)blockIdx.x * 4096;
  for (int i = tid; i < 4096; i += 256) {
    int r = i >> 6, cI = i & 63;
    Yp[i] = 0.5f * (Xm[r * 64 + cI] + Xm[cI * 64 + r]) + (r == cI ? 1e-3f : 0.0f);
  }
}

// ---------------------------------------------------------------------------
// MixedOp bimap: Y[k,b,co] = sym(sum_c W[k,co,c]·X[b,c]·W^T) + jitter, 30x30
// padded to 32x32 with identity padding.  KD = inner dim (64 for h, 32 for s2).
// Double-buffered: async-prefetch channel c+1 while WMMAs consume channel c.
// ---------------------------------------------------------------------------
template <int CI, int KD>
__global__ void __launch_bounds__(128)
bimap_mixed_kernel(const float* __restrict__ X, const float* __restrict__ W,
                   float* __restrict__ Y, int Co, int NB) {
  constexpr int WK = (KD == 64) ? 64 : 30;  // real weight column count
  constexpr int TPW = (KD == 64) ? 2 : 1;   // stage-1 tiles per wave
  constexpr int TPR = KD / 16;              // stage-1 tiles per row
  __shared__ float Wm[2][32 * KD];
  __shared__ float Xm[2][KD * KD];
  __shared__ float Tm[32 * KD];
  const int blk = blockIdx.x;
  const int k = blk / NB, bi = blk % NB;
  const int b = bi / Co, co = bi % Co;
  const int tid = threadIdx.x, wave = tid >> 5, lane = tid & 31;
  const int ym0 = (wave >> 1) * 16, yn0 = (wave & 1) * 16;
  v8f acc = {};

  // pre-zero W pad rows (30,31) of both buffers; async copies never touch them
  if constexpr (KD == 64) {
    for (int i = tid; i < 256; i += 128) {
      int bsel = i >> 7, j = i & 127;
      Wm[bsel][30 * 64 + j] = 0.0f;
    }
  }

#define PREFETCH(ci, bsel)                                                    \
  do {                                                                        \
    const float* Wp = W + (size_t)((k * Co + co) * CI + (ci)) * (30 * WK);    \
    const float* Xp = X + (size_t)(b * CI + (ci)) * (KD * KD);                \
    for (int i = tid; i < (KD * KD) / 4; i += 128)                            \
      async_ld_b128(&Xm[bsel][i * 4], Xp + i * 4);                            \
    if constexpr (KD == 64) {                                                 \
      for (int i = tid; i < 480; i += 128) /* rows 0..29, 16 chunks/row */    \
        async_ld_b128(&Wm[bsel][(i >> 4) * 64 + (i & 15) * 4], Wp + i * 4);   \
    } else { /* 30x30 unaligned rows: synchronous padded scatter */           \
      for (int i = tid; i < 1024; i += 128) {                                 \
        int r = i >> 5, cc = i & 31;                                          \
        Wm[bsel][i] = (r < 30 && cc < 30) ? Wp[r * 30 + cc] : 0.0f;           \
      }                                                                       \
    }                                                                         \
  } while (0)

  PREFETCH(0, 0);
  wait_async0();
  __syncthreads();
  for (int c = 0; c < CI; ++c) {
    const int cur = c & 1;
    if (c + 1 < CI) PREFETCH(c + 1, cur ^ 1);
    // stage 1: T(32xKD) = W * X
#pragma unroll
    for (int t = 0; t < TPW; ++t) {
      int tile = wave + 4 * t;
      int m0 = (tile / TPR) * 16, n0 = (tile % TPR) * 16;
      v8f cc = {};
#pragma unroll
      for (int k0 = 0; k0 < KD; k0 += 4)
        cc = wmma_k4(ldsA(Wm[cur], KD, m0, k0, lane),
                     ldsB(Xm[cur], KD, k0, n0, lane), cc);
      stC(Tm, KD, m0, n0, lane, cc);
    }
    __syncthreads();
    // stage 2: acc(32x32) += T * W^T
#pragma unroll
    for (int k0 = 0; k0 < KD; k0 += 4)
      acc = wmma_k4(ldsA(Tm, KD, ym0, k0, lane),
                    ldsBT(Wm[cur], KD, k0, yn0, lane), acc);
    wait_async0();   // prefetch of c+1 landed
    __syncthreads(); // everyone done with buffer cur and prefetch visible
  }
#undef PREFETCH
  stC(Xm[0], 32, ym0, yn0, lane, acc);  // Xm[0] reused as 32x32 staging
  __syncthreads();
  float* Yp = Y + (size_t)blk * 1024;
  for (int i = tid; i < 1024; i += 128) {
    int r = i >> 5, cc = i & 31;
    float v;
    if (r < 30 && cc < 30)
      v = 0.5f * (Xm[0][r * 32 + cc] + Xm[0][cc * 32 + r]) + (r == cc ? 1e-3f : 0.0f);
    else
      v = (r == cc) ? 1.0f : 0.0f;  // identity padding -> clean spectral maps
    Yp[i] = v;
  }
}

// ---------------------------------------------------------------------------
// Sandwich: O[blk] = sym(A[blk % amod] · S[blk] · A[blk % amod]) (A symmetric),
// 32x32.  Used for Gis·S·Gis and Gs·expm(T)·Gs in the Karcher step.
// ---------------------------------------------------------------------------
__global__ void __launch_bounds__(128)
sandwich_kernel(const float* __restrict__ A, const float* __restrict__ S,
                float* __restrict__ O, int amod) {
  __shared__ float Am[1024];
  __shared__ float Sm[1024];
  __shared__ float Tm[1024];
  const int blk = blockIdx.x, tid = threadIdx.x, wave = tid >> 5, lane = tid & 31;
  const float* Ap = A + (size_t)(blk % amod) * 1024;
  const float* Sp = S + (size_t)blk * 1024;
  for (int i = tid; i < 256; i += 128) {
    async_ld_b128(&Am[i * 4], Ap + i * 4);
    async_ld_b128(&Sm[i * 4], Sp + i * 4);
  }
  wait_async0();
  __syncthreads();
  const int m0 = (wave >> 1) * 16, n0 = (wave & 1) * 16;
  v8f cc = {};
#pragma unroll
  for (int k0 = 0; k0 < 32; k0 += 4)
    cc = wmma_k4(ldsA(Am, 32, m0, k0, lane), ldsB(Sm, 32, k0, n0, lane), cc);
  stC(Tm, 32, m0, n0, lane, cc);
  __syncthreads();
  v8f dd = {};
#pragma unroll
  for (int k0 = 0; k0 < 32; k0 += 4)
    dd = wmma_k4(ldsA(Tm, 32, m0, k0, lane), ldsBT(Am, 32, k0, n0, lane), dd);
  stC(Sm, 32, m0, n0, lane, dd);  // Sm free after stage-1 sync
  __syncthreads();
  float* Op = O + (size_t)blk * 1024;
  for (int i = tid; i < 1024; i += 128) {
    int r = i >> 5, c = i & 31;
    Op[i] = 0.5f * (Sm[r * 32 + c] + Sm[c * 32 + r]);
  }
}

// ---------------------------------------------------------------------------
// Parallel cyclic Jacobi eigensolver (workgroup-per-matrix, A & V in LDS)
// with fused spectral function + V·f(Λ)·Vᵀ reconstruction.
// mode 0: ReEig max(w,1e-4)   -> out1
// mode 1: sqrt(max(w,0))      -> out1 ; rsqrt(max(w,1e-10)) -> out2
// mode 2: log(max(w,1e-10))   -> out1
// mode 3: exp(w)              -> out1
// ---------------------------------------------------------------------------
DEVINL int tourn_player(int slot, int round, int n) {
  return (slot == 0) ? 0 : 1 + ((slot - 1 + round) % (n - 1));
}

template <int N>
__global__ void __launch_bounds__(N)
jacobi_kernel(const float* __restrict__ in, float* __restrict__ out1,
              float* __restrict__ out2, int mode) {
  constexpr int NP = N / 2;
  constexpr int NSWEEPS = 8;
  __shared__ float Am[N * N];
  __shared__ float Vm[N * N];
  __shared__ float rc[NP], rs[NP];
  __shared__ int rp[NP], rq[NP];
  __shared__ float fd1[N], fd2[N];
  const int tid = threadIdx.x;
  const float* Ain = in + (size_t)blockIdx.x * (N * N);
  for (int i = 0; i < N; ++i) {
    Am[i * N + tid] = Ain[i * N + tid];
    Vm[i * N + tid] = (i == tid) ? 1.0f : 0.0f;
  }
  __syncthreads();
  for (int sweep = 0; sweep < NSWEEPS; ++sweep) {
    for (int round = 0; round < N - 1; ++round) {
      if (tid < NP) {
        int p = tourn_player(tid, round, N);
        int q = tourn_player(N - 1 - tid, round, N);
        if (p > q) { int t = p; p = q; q = t; }
        float app = Am[p * N + p], aqq = Am[q * N + q], apq = Am[p * N + q];
        float c = 1.0f, s = 0.0f;
        if (fabsf(apq) > 1e-12f) {
          float tau = (aqq - app) / (2.0f * apq);
          float t = ((tau >= 0.0f) ? 1.0f : -1.0f) /
                    (fabsf(tau) + sqrtf(1.0f + tau * tau));
          c = rsqrtf(1.0f + t * t);
          s = t * c;
        }
        rc[tid] = c; rs[tid] = s; rp[tid] = p; rq[tid] = q;
      }
      __syncthreads();
      // A <- A·J ; V <- V·J  (thread tid owns row tid)
      for (int pr = 0; pr < NP; ++pr) {
        int p = rp[pr], q = rq[pr];
        float c = rc[pr], s = rs[pr];
        float ap = Am[tid * N + p], aq = Am[tid * N + q];
        Am[tid * N + p] = c * ap - s * aq;
        Am[tid * N + q] = s * ap + c * aq;
        float vp = Vm[tid * N + p], vq = Vm[tid * N + q];
        Vm[tid * N + p] = c * vp - s * vq;
        Vm[tid * N + q] = s * vp + c * vq;
      }
      __syncthreads();
      // A <- Jᵀ·A  (thread tid owns column tid)
      for (int pr = 0; pr < NP; ++pr) {
        int p = rp[pr], q = rq[pr];
        float c = rc[pr], s = rs[pr];
        float ap = Am[p * N + tid], aq = Am[q * N + tid];
        Am[p * N + tid] = c * ap - s * aq;
        Am[q * N + tid] = s * ap + c * aq;
      }
      __syncthreads();
    }
  }
  // spectral function on eigenvalues (diagonal)
  {
    float d = Am[tid * N + tid];
    float f1, f2 = 0.0f;
    if (mode == 0)      f1 = fmaxf(d, 1e-4f);
    else if (mode == 1) { f1 = sqrtf(fmaxf(d, 0.0f)); f2 = rsqrtf(fmaxf(d, 1e-10f)); }
    else if (mode == 2) f1 = logf(fmaxf(d, 1e-10f));
    else                f1 = expf(d);
    fd1[tid] = f1; fd2[tid] = f2;
  }
  __syncthreads();
  // out = V diag(f) Vᵀ  (thread tid owns column tid)
  float* O1 = out1 + (size_t)blockIdx.x * (N * N);
  for (int i = 0; i < N; ++i) {
    float acc = 0.0f;
    for (int kk = 0; kk < N; ++kk) acc += Vm[i * N + kk] * fd1[kk] * Vm[tid * N + kk];
    O1[i * N + tid] = acc;
  }
  if (out2 != nullptr) {
    float* O2 = out2 + (size_t)blockIdx.x * (N * N);
    for (int i = 0; i < N; ++i) {
      float acc = 0.0f;
      for (int kk = 0; kk < N; ++kk) acc += Vm[i * N + kk] * fd2[kk] * Vm[tid * N + kk];
      O2[i * N + tid] = acc;
    }
  }
}

// ---------------------------------------------------------------------------
// Weighted sum over K stacked matrix slabs: G[bi] = sum_k w[k] * S[k][bi].
// ---------------------------------------------------------------------------
__global__ void __launch_bounds__(128)
wsum_kernel(const float* __restrict__ S, const float* __restrict__ wv,
            float* __restrict__ G, int K, long strideK) {
  const int bi = blockIdx.x, tid = threadIdx.x;
  const float* base = S + (size_t)bi * 1024;
  for (int i = tid; i < 1024; i += 128) {
    float acc = 0.0f;
    for (int k = 0; k < K; ++k) acc += wv[k] * base[(size_t)k * strideK + i];
    G[(size_t)bi * 1024 + i] = acc;
  }
}

// softmax of the 5x4 arch weights + the uniform state-barycenter weights
__global__ void __launch_bounds__(32)
prep_weights_kernel(const float* __restrict__ w_in, float* __restrict__ w_out) {
  const int tid = threadIdx.x;
  if (tid < 5) {
    float a = w_in[tid * 4 + 0], b = w_in[tid * 4 + 1];
    float c = w_in[tid * 4 + 2], d = w_in[tid * 4 + 3];
    float m = fmaxf(fmaxf(a, b), fmaxf(c, d));
    float ea = expf(a - m), eb = expf(b - m), ec = expf(c - m), ed = expf(d - m);
    float s = 1.0f / (ea + eb + ec + ed);
    w_out[tid * 4 + 0] = ea * s; w_out[tid * 4 + 1] = eb * s;
    w_out[tid * 4 + 2] = ec * s; w_out[tid * 4 + 3] = ed * s;
  }
  if (tid == 5) {
    w_out[20] = 0.5f; w_out[21] = 0.5f;
    w_out[22] = 1.0f / 3.0f; w_out[23] = 1.0f / 3.0f; w_out[24] = 1.0f / 3.0f;
  }
}

// gather padded 32x32 states into the [B,2C,30,30] output
__global__ void __launch_bounds__(256)
gather_out_kernel(const float* __restrict__ s2, const float* __restrict__ s3,
                  float* __restrict__ out, int n) {
  int idx = blockIdx.x * blockDim.x + threadIdx.x;
  if (idx >= n) return;
  int rc = idx % 900;
  int ch = (idx / 900) % 16;
  int b = idx / (900 * 16);
  int r = rc / 30, c = rc % 30;
  const float* src = (ch < 8) ? s2 : s3;
  out[idx] = src[(size_t)(b * 8 + (ch & 7)) * 1024 + r * 32 + c];
}

// ---------------------------------------------------------------------------
// Host-side orchestration
// ---------------------------------------------------------------------------
static void run_bary(float* S, int K, int NB, const float* wv, float* G,
                     float* Gs, float* Gis, float* T, float* dst,
                     hipStream_t stream) {
  wsum_kernel<<<NB, 128, 0, stream>>>(S, wv, G, K, (long)NB * 1024);
  jacobi_kernel<32><<<NB, 32, 0, stream>>>(G, Gs, Gis, 1);          // sqrtm+isqrtm
  sandwich_kernel<<<K * NB, 128, 0, stream>>>(Gis, S, S, NB);       // Gis·S·Gis
  jacobi_kernel<32><<<K * NB, 32, 0, stream>>>(S, S, nullptr, 2);   // logm
  wsum_kernel<<<NB, 128, 0, stream>>>(S, wv, T, K, (long)NB * 1024);
  jacobi_kernel<32><<<NB, 32, 0, stream>>>(T, T, nullptr, 3);       // expm
  sandwich_kernel<<<NB, 128, 0, stream>>>(Gs, T, dst, NB);          // Gs·E·Gs
}

static void run_mixed(const float* X, const float* We, const float* wv,
                      float* outs, float* G, float* Gs, float* Gis, float* T,
                      float* dst, int edge4, hipStream_t stream) {
  const int NB = 1024;  // B*Co = 128*8
  if (edge4)
    bimap_mixed_kernel<8, 32><<<4 * NB, 128, 0, stream>>>(X, We, outs, 8, NB);
  else
    bimap_mixed_kernel<16, 64><<<4 * NB, 128, 0, stream>>>(X, We, outs, 8, NB);
  jacobi_kernel<32><<<4 * NB, 32, 0, stream>>>(outs, outs, nullptr, 0);  // ReEig
  run_bary(outs, 4, NB, wv, G, Gs, Gis, T, dst, stream);
}

extern "C" void kernel_launch(void* const* d_in, const int* in_sizes, int n_in,
                              void* d_out, int out_size, void* d_ws,
                              size_t ws_size, hipStream_t stream) {
  (void)in_sizes; (void)n_in; (void)out_size; (void)ws_size;
  const float* s0  = (const float*)d_in[0];
  const float* s1  = (const float*)d_in[1];
  const float* wts = (const float*)d_in[2];
  const float* Wp0 = (const float*)d_in[3];
  const float* Wp1 = (const float*)d_in[4];
  const float* We[5] = {(const float*)d_in[5], (const float*)d_in[6],
                        (const float*)d_in[7], (const float*)d_in[8],
                        (const float*)d_in[9]};
  float* ws = (float*)d_ws;
  size_t off = 0;
  float* h0   = ws + off; off += (size_t)2048 * 4096;  // [B,2C,64,64]
  float* h1   = ws + off; off += (size_t)2048 * 4096;
  float* outs = ws + off; off += (size_t)4 * 1024 * 1024;  // [K][B*Co][32*32]
  float* mst  = ws + off; off += (size_t)3 * 1024 * 1024;  // stacked edge outs
  float* G    = ws + off; off += (size_t)1024 * 1024;
  float* Gs   = ws + off; off += (size_t)1024 * 1024;
  float* Gis  = ws + off; off += (size_t)1024 * 1024;
  float* T    = ws + off; off += (size_t)1024 * 1024;
  float* s2b  = ws + off; off += (size_t)1024 * 1024;  // states[2], padded
  float* s3b  = ws + off; off += (size_t)1024 * 1024;  // states[3], padded
  float* wbuf = ws + off; off += 32;

  prep_weights_kernel<<<1, 32, 0, stream>>>(wts, wbuf);

  // preprocess0/1: bimap + ReEig, n = 64
  bimap64_kernel<8><<<2048, 256, 0, stream>>>(s0, Wp0, h0, 16);
  jacobi_kernel<64><<<2048, 64, 0, stream>>>(h0, h0, nullptr, 0);
  bimap64_kernel<16><<<2048, 256, 0, stream>>>(s1, Wp1, h1, 16);
  jacobi_kernel<64><<<2048, 64, 0, stream>>>(h1, h1, nullptr, 0);

  const size_t MSLAB = (size_t)1024 * 1024;
  // step 0: edges 0,1 then uniform barycenter -> states[2]
  run_mixed(h0, We[0], wbuf + 0, outs, G, Gs, Gis, T, mst + 0 * MSLAB, 0, stream);
  run_mixed(h1, We[1], wbuf + 4, outs, G, Gs, Gis, T, mst + 1 * MSLAB, 0, stream);
  run_bary(mst, 2, 1024, wbuf + 20, G, Gs, Gis, T, s2b, stream);
  // step 1: edges 2,3,4 then uniform barycenter -> states[3]
  run_mixed(h0,  We[2], wbuf + 8,  outs, G, Gs, Gis, T, mst + 0 * MSLAB, 0, stream);
  run_mixed(h1,  We[3], wbuf + 12, outs, G, Gs, Gis, T, mst + 1 * MSLAB, 0, stream);
  run_mixed(s2b, We[4], wbuf + 16, outs, G, Gs, Gis, T, mst + 2 * MSLAB, 1, stream);
  run_bary(mst, 3, 1024, wbuf + 22, G, Gs, Gis, T, s3b, stream);

  const int n = 128 * 16 * 900;
  gather_out_kernel<<<(n + 255) / 256, 256, 0, stream>>>(s2b, s3b, (float*)d_out, n);
}